// GAT_68556267979246
// MI455X (gfx1250) — compile-verified
//
#include <hip/hip_runtime.h>
#include <hip/hip_bf16.h>

typedef float v2f __attribute__((ext_vector_type(2)));
typedef float v8f __attribute__((ext_vector_type(8)));

#define GAT_N      50000
#define GAT_E      800000
#define GAT_INCH   128
#define GAT_HID    32
#define GAT_HEADS  4
#define GAT_OUTCH  64
#define GAT_SLOPE  0.2f

// ---------------------------------------------------------------------------
// f32 <-> monotone-sortable u32 key (for exact float atomicMax via u32 max)
// ---------------------------------------------------------------------------
__device__ __forceinline__ unsigned gat_f2key(float f) {
    unsigned b = __float_as_uint(f);
    return (b & 0x80000000u) ? ~b : (b | 0x80000000u);
}
__device__ __forceinline__ float gat_key2f(unsigned k) {
    return __uint_as_float((k & 0x80000000u) ? (k & 0x7FFFFFFFu) : ~k);
}

__device__ __forceinline__ void gat_edge(const long long* ei, int e, int E,
                                         int& s, int& d) {
    if (e < E) { s = (int)ei[e]; d = (int)ei[E + e]; }
    else       { s = d = e - E; }               // self loop
}

// ---------------------------------------------------------------------------
// WMMA f32 GEMM: C[rows x NC] = A[rows x K] @ W[K x NC], rows % 16 == 0,
// K % 4 == 0.  One block = 16 rows, NC/16 waves, each wave one 16x16 tile.
// Uses V_WMMA_F32_16X16X4_F32 (full fp32 precision on the matrix pipe).
// ---------------------------------------------------------------------------
template <int NC>
__global__ void __launch_bounds__(32 * (NC / 16))
gat_gemm16_wmma(const float* __restrict__ A, const float* __restrict__ W,
                float* __restrict__ C, int K) {
    const int lane = threadIdx.x & 31;
    const int wave = threadIdx.x >> 5;
    const int m    = lane & 15;       // A-row in tile / B,C column in tile
    const int half = lane >> 4;       // K-pair select (A/B), row-half (C/D)
    const int row0 = blockIdx.x * 16;
    const int col  = wave * 16 + m;

    const float* __restrict__ arow = A + (long long)(row0 + m) * K + 2 * half;

    v8f c = {};
#pragma unroll 4
    for (int k = 0; k < K; k += 4) {
        v2f a, b;
        a.x = arow[k];
        a.y = arow[k + 1];
        b.x = W[(long long)(k + 2 * half)     * NC + col];
        b.y = W[(long long)(k + 2 * half + 1) * NC + col];
        c = __builtin_amdgcn_wmma_f32_16x16x4_f32(
                /*neg_a=*/false, a, /*neg_b=*/false, b,
                /*c_mod=*/(short)0, c, /*reuse_a=*/false, /*reuse_b=*/false);
    }
#pragma unroll
    for (int v = 0; v < 8; ++v)
        C[(long long)(row0 + 8 * half + v) * NC + col] = c[v];
}

// ---------------------------------------------------------------------------
// Per-node attention scalars: a_src[n,h] = dot(h[n,h,:], att_src[h,:])
// ---------------------------------------------------------------------------
template <int H, int CH>
__global__ void gat_attn_scalars(const float* __restrict__ hbuf,
                                 const float* __restrict__ att_s,
                                 const float* __restrict__ att_d,
                                 float* __restrict__ a_s,
                                 float* __restrict__ a_d, int nNodes) {
    int t = blockIdx.x * blockDim.x + threadIdx.x;
    if (t >= nNodes * H) return;
    int n = t / H, h = t % H;
    const float* hp = hbuf + (long long)n * H * CH + h * CH;
    const float* as = att_s + h * CH;
    const float* ad = att_d + h * CH;
    float ss = 0.f, sd = 0.f;
#pragma unroll 8
    for (int c = 0; c < CH; ++c) {
        float v = hp[c];
        ss += v * as[c];
        sd += v * ad[c];
    }
    a_s[t] = ss;
    a_d[t] = sd;
}

// ---------------------------------------------------------------------------
// Fills (harness does not re-zero scratch between replays)
// ---------------------------------------------------------------------------
__global__ void gat_fill_f32(float* p, float v, int n) {
    int t = blockIdx.x * blockDim.x + threadIdx.x;
    if (t < n) p[t] = v;
}
__global__ void gat_fill_u32(unsigned* p, unsigned v, int n) {
    int t = blockIdx.x * blockDim.x + threadIdx.x;
    if (t < n) p[t] = v;
}

// ---------------------------------------------------------------------------
// Edge pass A: alpha = leakyrelu(a_src[s]+a_dst[d]); segment max via u32 key
// ---------------------------------------------------------------------------
template <int H>
__global__ void gat_edge_alpha(const long long* __restrict__ ei,
                               const float* __restrict__ a_s,
                               const float* __restrict__ a_d,
                               float* __restrict__ alpha,
                               unsigned* __restrict__ amaxk,
                               int E, int nNodes) {
    int t = blockIdx.x * blockDim.x + threadIdx.x;
    if (t >= (E + nNodes) * H) return;
    int e = t / H, h = t % H, s, d;
    gat_edge(ei, e, E, s, d);
    float a = a_s[s * H + h] + a_d[d * H + h];
    a = a > 0.f ? a : GAT_SLOPE * a;
    alpha[t] = a;
    atomicMax(&amaxk[d * H + h], gat_f2key(a));
}

// ---------------------------------------------------------------------------
// Edge pass B: ex = exp(alpha - amax[dst]); denom[dst] += ex
// ---------------------------------------------------------------------------
template <int H>
__global__ void gat_edge_exp(const long long* __restrict__ ei,
                             float* __restrict__ alpha,
                             const unsigned* __restrict__ amaxk,
                             float* __restrict__ denom, int E, int nNodes) {
    int t = blockIdx.x * blockDim.x + threadIdx.x;
    if (t >= (E + nNodes) * H) return;
    int e = t / H, h = t % H, s, d;
    gat_edge(ei, e, E, s, d);
    float mx = gat_key2f(amaxk[d * H + h]);
    float ex = __expf(alpha[t] - mx);
    alpha[t] = ex;
    atomicAdd(&denom[d * H + h], ex);
}

// ---------------------------------------------------------------------------
// Edge pass C: accum[dst,h,:] += h[src,h,:] * (ex / (denom[dst,h]+eps))
// ---------------------------------------------------------------------------
template <int H, int CH>
__global__ void gat_edge_msg(const long long* __restrict__ ei,
                             const float* __restrict__ hbuf,
                             const float* __restrict__ ex,
                             const float* __restrict__ denom,
                             float* __restrict__ accum, int E, int nNodes) {
    int t = blockIdx.x * blockDim.x + threadIdx.x;
    if (t >= (E + nNodes) * H) return;
    int e = t / H, h = t % H, s, d;
    gat_edge(ei, e, E, s, d);
    float att = ex[t] / (denom[d * H + h] + 1e-16f);
    const float* hs = hbuf + (long long)s * H * CH + h * CH;
    float* acc      = accum + (long long)d * H * CH + h * CH;
#pragma unroll 4
    for (int c = 0; c < CH; ++c)
        atomicAdd(&acc[c], hs[c] * att);
}

// ---------------------------------------------------------------------------
// Elementwise epilogues
// ---------------------------------------------------------------------------
__global__ void gat_bias_lrelu(float* buf, const float* __restrict__ bias,
                               int total, int width) {
    int t = blockIdx.x * blockDim.x + threadIdx.x;
    if (t >= total) return;
    float v = buf[t] + bias[t % width];
    buf[t] = v > 0.f ? v : GAT_SLOPE * v;
}
__global__ void gat_bias_add(float* buf, const float* __restrict__ bias,
                             int total, int width) {
    int t = blockIdx.x * blockDim.x + threadIdx.x;
    if (t >= total) return;
    buf[t] += bias[t % width];
}

// ---------------------------------------------------------------------------
extern "C" void kernel_launch(void* const* d_in, const int* in_sizes, int n_in,
                              void* d_out, int out_size, void* d_ws,
                              size_t ws_size, hipStream_t stream) {
    (void)in_sizes; (void)n_in; (void)out_size; (void)ws_size;
    const int N = GAT_N, E = GAT_E;
    const float*     x        = (const float*)d_in[0];
    const long long* ei       = (const long long*)d_in[1];
    const float*     W1       = (const float*)d_in[2];
    const float*     att_src1 = (const float*)d_in[3];
    const float*     att_dst1 = (const float*)d_in[4];
    const float*     b1       = (const float*)d_in[5];
    const float*     W2       = (const float*)d_in[6];
    const float*     att_src2 = (const float*)d_in[7];
    const float*     att_dst2 = (const float*)d_in[8];
    const float*     b2       = (const float*)d_in[9];
    float*           out      = (float*)d_out;

    // workspace layout (floats)
    float*    h1    = (float*)d_ws;                       // N*128 (layer2 reuses for h2: N*64)
    float*    acc1  = h1 + (size_t)N * 128;               // N*128 (becomes hmid in place)
    float*    a_s   = acc1 + (size_t)N * 128;             // N*4
    float*    a_d   = a_s + (size_t)N * 4;                // N*4
    unsigned* amaxk = (unsigned*)(a_d + (size_t)N * 4);   // N*4
    float*    denom = (float*)(amaxk + (size_t)N * 4);    // N*4
    float*    alpha = denom + (size_t)N * 4;              // (E+N)*4

    const int T = 256;
    const int EH4 = (E + N) * 4, EH1 = (E + N);
    auto blk = [](int n, int t) { return (n + t - 1) / t; };

    // ---------------- Layer 1 (4 heads x 32 ch, concat) ----------------
    gat_gemm16_wmma<128><<<N / 16, 256, 0, stream>>>(x, W1, h1, 128);
    gat_attn_scalars<4, 32><<<blk(N * 4, T), T, 0, stream>>>(h1, att_src1, att_dst1, a_s, a_d, N);
    gat_fill_u32<<<blk(N * 4, T), T, 0, stream>>>(amaxk, 0u, N * 4);
    gat_fill_f32<<<blk(N * 4, T), T, 0, stream>>>(denom, 0.f, N * 4);
    gat_fill_f32<<<blk(N * 128, T), T, 0, stream>>>(acc1, 0.f, N * 128);
    gat_edge_alpha<4><<<blk(EH4, T), T, 0, stream>>>(ei, a_s, a_d, alpha, amaxk, E, N);
    gat_edge_exp<4><<<blk(EH4, T), T, 0, stream>>>(ei, alpha, amaxk, denom, E, N);
    gat_edge_msg<4, 32><<<blk(EH4, T), T, 0, stream>>>(ei, h1, alpha, denom, acc1, E, N);
    gat_bias_lrelu<<<blk(N * 128, T), T, 0, stream>>>(acc1, b1, N * 128, 128);

    // ---------------- Layer 2 (1 head x 64 ch, mean==identity) ----------------
    gat_gemm16_wmma<64><<<N / 16, 128, 0, stream>>>(acc1, W2, h1, 128);
    gat_attn_scalars<1, 64><<<blk(N, T), T, 0, stream>>>(h1, att_src2, att_dst2, a_s, a_d, N);
    gat_fill_u32<<<blk(N, T), T, 0, stream>>>(amaxk, 0u, N);
    gat_fill_f32<<<blk(N, T), T, 0, stream>>>(denom, 0.f, N);
    gat_fill_f32<<<blk(N * 64, T), T, 0, stream>>>(out, 0.f, N * 64);
    gat_edge_alpha<1><<<blk(EH1, T), T, 0, stream>>>(ei, a_s, a_d, alpha, amaxk, E, N);
    gat_edge_exp<1><<<blk(EH1, T), T, 0, stream>>>(ei, alpha, amaxk, denom, E, N);
    gat_edge_msg<1, 64><<<blk(EH1, T), T, 0, stream>>>(ei, h1, alpha, denom, out, E, N);
    gat_bias_add<<<blk(N * 64, T), T, 0, stream>>>(out, b2, N * 64, 64);
}